// JavascriptExtractor_35527969472765
// MI455X (gfx1250) — compile-verified
//
#include <hip/hip_runtime.h>
#include <hip/hip_bf16.h>
#include <stdint.h>

// Problem constants (from the reference)
#define BATCH 32
#define S_SRC 2048
#define S_Q   512
#define LPACK (S_SRC + S_Q)   // 2560 packed positions per row
#define D_EMB 1024
#define D4    (D_EMB / 4)     // 256 float4 per embedding row (4 KB)

typedef float __attribute__((ext_vector_type(4))) v4f;
typedef int   __attribute__((ext_vector_type(4))) v4i;

// ---------------------------------------------------------------------------
// Kernel 1: per-row nonzero counts (sl, ql). 32 blocks, one per batch row.
// Vectorized: each thread does two b128 loads from the source row
// (512 int4 / 256 threads) and threads < 128 one b128 load from the query
// row (128 int4) -- independent wide loads instead of 8 dependent b32
// load/wait round trips. Writes int2 {sl, ql} into workspace.
// ---------------------------------------------------------------------------
__global__ __launch_bounds__(256) void ragged_len_kernel(
    const int* __restrict__ src, const int* __restrict__ qry,
    int2* __restrict__ lens) {
  __shared__ int sbuf[256];
  __shared__ int qbuf[256];
  const int row = blockIdx.x;
  const v4i* s4 = reinterpret_cast<const v4i*>(src + row * S_SRC);  // 512 v4i
  const v4i* q4 = reinterpret_cast<const v4i*>(qry + row * S_Q);    // 128 v4i

  const v4i a = s4[threadIdx.x];
  const v4i b = s4[threadIdx.x + 256];
  int cs = (a.x != 0) + (a.y != 0) + (a.z != 0) + (a.w != 0) +
           (b.x != 0) + (b.y != 0) + (b.z != 0) + (b.w != 0);
  int cq = 0;
  if (threadIdx.x < 128) {
    const v4i c = q4[threadIdx.x];
    cq = (c.x != 0) + (c.y != 0) + (c.z != 0) + (c.w != 0);
  }

  sbuf[threadIdx.x] = cs;
  qbuf[threadIdx.x] = cq;
  __syncthreads();
  for (int o = 128; o > 0; o >>= 1) {
    if ((int)threadIdx.x < o) {
      sbuf[threadIdx.x] += sbuf[threadIdx.x + o];
      qbuf[threadIdx.x] += qbuf[threadIdx.x + o];
    }
    __syncthreads();
  }
  if (threadIdx.x == 0) lens[row] = make_int2(sbuf[0], qbuf[0]);
}

// ---------------------------------------------------------------------------
// Kernel 2: gather. One wave32 per packed token (B*L = 81920 tokens).
// Block = 256 threads = 8 waves; grid = 81920/8 = 10240 (exact, EXEC all-1s).
//
// Pure async-DMA copy per token (no VGPR data staging at all):
//   - 8 x global_load_async_to_lds_b128   (emb row -> LDS, RT: let the
//     ~206 MB table live in the 192 MB L2; avg row reuse ~1.6x)
//   - s_wait_asynccnt 0                   (async loads complete in order)
//   - 8 x global_store_async_from_lds_b128 th:TH_STORE_NT (LDS -> out,
//     write-once stream: don't pollute L2)
//   - s_endpgm's implicit wait-idle drains the async stores.
// 8 KB LDS/block -> many resident waves/SIMD to hide the
// asynccnt stall.
// ---------------------------------------------------------------------------
__global__ __launch_bounds__(256) void gather_kernel(
    const int* __restrict__ src, const int* __restrict__ qry,
    const float* __restrict__ emb, const int2* __restrict__ lens,
    float* __restrict__ out) {
  __shared__ v4f buf[8 * 64];  // 8 waves x 64 float4 = 1 KB slot per wave (8 KB)

  const int wave = threadIdx.x >> 5;
  const int lane = threadIdx.x & 31;
  const int tok  = blockIdx.x * 8 + wave;   // 0 .. 81919 (uniform per wave)
  const int row  = tok / LPACK;
  const int pos  = tok - row * LPACK;

  const int2 sq = lens[row];                // {sl, ql}
  int token;
  if (pos < sq.x) {
    token = src[row * S_SRC + pos];
  } else {
    const int qp = pos - sq.x;
    token = (qp < sq.y) ? qry[row * S_Q + qp] : 0;  // pad -> emb row 0
  }

  const v4f* __restrict__ rowp =
      reinterpret_cast<const v4f*>(emb) + (size_t)token * D4;

  // Generic->LDS offset: addrspacecast(local->flat) keeps the LDS byte offset
  // in the low 32 bits of the flat address, so truncation yields the DS addr.
  const uint32_t l0 = (uint32_t)(uintptr_t)(&buf[wave * 64]) +
                      (uint32_t)(lane * 16);
  const v4f* g0 = rowp + lane;
  v4f* o0 = reinterpret_cast<v4f*>(out) + (size_t)tok * D4 + lane;

  // 8 async wave-wide 512 B transfers = one 4 KB embedding row into LDS.
#pragma unroll
  for (int k = 0; k < 8; ++k) {
    const v4f* g = g0 + k * 32;
    uint32_t l = l0 + (uint32_t)(k * 512);
    asm volatile("global_load_async_to_lds_b128 %0, %1, off"
                 :: "v"(l), "v"(g)
                 : "memory");
  }
  asm volatile("s_wait_asynccnt 0" ::: "memory");

  // 8 async wave-wide 512 B transfers: LDS -> output, non-temporal.
#pragma unroll
  for (int k = 0; k < 8; ++k) {
    v4f* o = o0 + k * 32;
    uint32_t l = l0 + (uint32_t)(k * 512);
    asm volatile("global_store_async_from_lds_b128 %0, %1, off th:TH_STORE_NT"
                 :: "v"(o), "v"(l)
                 : "memory");
  }
  // s_endpgm performs an implicit wait-idle: async stores drain before exit.
}

// ---------------------------------------------------------------------------
// Launcher. Inputs (setup_inputs order): sources[i32 B*S_SRC],
// queries[i32 B*S_Q], emb[f32 VOCAB*D_EMB]. Output: f32 B*LPACK*D_EMB.
// ---------------------------------------------------------------------------
extern "C" void kernel_launch(void* const* d_in, const int* in_sizes, int n_in,
                              void* d_out, int out_size, void* d_ws, size_t ws_size,
                              hipStream_t stream) {
  (void)in_sizes; (void)n_in; (void)out_size; (void)ws_size;
  const int*   sources = (const int*)d_in[0];
  const int*   queries = (const int*)d_in[1];
  const float* emb     = (const float*)d_in[2];
  float*       out     = (float*)d_out;
  int2*        lens    = (int2*)d_ws;      // 32 * 8 B scratch

  ragged_len_kernel<<<BATCH, 256, 0, stream>>>(sources, queries, lens);
  gather_kernel<<<(BATCH * LPACK) / 8, 256, 0, stream>>>(sources, queries, emb,
                                                         lens, out);
}